// GCN_2422361555109
// MI455X (gfx1250) — compile-verified
//
#include <hip/hip_runtime.h>
#include <math.h>

typedef __attribute__((ext_vector_type(2))) float v2f;
typedef __attribute__((ext_vector_type(8))) float v8f;

#define NODE_DIM_IN 128

// ---------------------------------------------------------------------------
// Degree / normalization precompute (shared by all 3 layers)
// ---------------------------------------------------------------------------
__global__ void deg_init_kernel(float* __restrict__ deg, int n) {
    int i = blockIdx.x * blockDim.x + threadIdx.x;
    if (i < n) deg[i] = 1.0f;  // self-loop weight
}

__global__ void deg_acc_kernel(const int* __restrict__ col,
                               const float* __restrict__ ew,
                               float* __restrict__ deg, int E) {
    int e = blockIdx.x * blockDim.x + threadIdx.x;
    if (e < E) atomicAdd(&deg[col[e]], ew[e]);
}

// deg -> dis (in place), selfnorm = dis*dis
__global__ void dinv_kernel(float* __restrict__ deg,
                            float* __restrict__ selfn, int n) {
    int i = blockIdx.x * blockDim.x + threadIdx.x;
    if (i >= n) return;
    float d = deg[i];
    float dis = (d > 0.0f) ? rsqrtf(d) : 0.0f;
    deg[i] = dis;
    selfn[i] = dis * dis;
}

__global__ void edge_norm_kernel(const int* __restrict__ row,
                                 const int* __restrict__ col,
                                 const float* __restrict__ ew,
                                 const float* __restrict__ dis,
                                 float* __restrict__ nrm, int E) {
    int e = blockIdx.x * blockDim.x + threadIdx.x;
    if (e >= E) return;
    nrm[e] = dis[row[e]] * ew[e] * dis[col[e]];
}

// ---------------------------------------------------------------------------
// WMMA f32 GEMM + fused self-loop init:
//   H[n,F]   = A[n,K] @ W[K,F]
//   AGG[n,F] = selfn[row] * H
// block = 128 threads = 4 waves; each wave computes one 16-row tile.
// Branch-free inner loop: row index clamped (store-guarded), A k-overrun is
// multiplied by the zero-padded B tile in LDS, so no per-element guards.
// ---------------------------------------------------------------------------
template <int K, int F>
__global__ void __launch_bounds__(128)
wmma_gemm_kernel(const float* __restrict__ A, const float* __restrict__ W,
                 const float* __restrict__ selfn,
                 float* __restrict__ H, float* __restrict__ AGG, int n) {
    constexpr int Kp = (K + 3) & ~3;
    __shared__ float sW[Kp * 16];             // [Kp][16], zero-padded

    for (int idx = threadIdx.x; idx < Kp * 16; idx += 128) {
        int k = idx >> 4, f = idx & 15;
        sW[idx] = (k < K && f < F) ? W[k * F + f] : 0.0f;
    }
    __syncthreads();

    const int wave = threadIdx.x >> 5;        // wave32
    const int lane = threadIdx.x & 31;
    const int m    = lane & 15;               // A row within tile / B,D column
    const int kb   = (lane >> 4) << 1;        // 0 or 2

    const long tile_row0 = ((long)blockIdx.x * 4 + wave) * 16;
    long row = tile_row0 + m;
    if (row >= n) row = n - 1;                // clamp; duplicates discarded at store
    const float* Arow = A + row * (long)K;

    v8f acc = {};
#pragma unroll
    for (int k0 = 0; k0 < Kp; k0 += 4) {
        const int k = k0 + kb;
        v2f a = *(const v2f*)(Arow + k);      // 8B-aligned (k even)
        v2f b;
        b.x = sW[k * 16 + m];
        b.y = sW[(k + 1) * 16 + m];
        acc = __builtin_amdgcn_wmma_f32_16x16x4_f32(
            false, a, false, b, (short)0, acc, false, false);
    }

    // C/D layout: VGPR r -> row = r + 8*(lane>=16), col = lane&15
    const long rbase = tile_row0 + 8 * (lane >> 4);
    if (m < F) {
#pragma unroll
        for (int r = 0; r < 8; ++r) {
            long rr = rbase + r;
            if (rr < n) {
                float v = acc[r];
                H[rr * (long)F + m]   = v;
                AGG[rr * (long)F + m] = selfn[rr] * v;
            }
        }
    }
}

// ---------------------------------------------------------------------------
// Edge scatter: AGG[col] += norm[e] * H[row]   (vectorized feature gather)
// ---------------------------------------------------------------------------
template <int F>
__global__ void scatter_kernel(const int* __restrict__ row,
                               const int* __restrict__ col,
                               const float* __restrict__ nrm,
                               const float* __restrict__ h,
                               float* __restrict__ out, int E) {
    int e = blockIdx.x * blockDim.x + threadIdx.x;
    if (e >= E) return;
    const int r = row[e], c = col[e];
    const float w = nrm[e];
    const float* hr = h + (long)r * F;
    float* oc = out + (long)c * F;

    float hv[F];
    if constexpr ((F & 3) == 0) {             // 16B-aligned rows (F=12)
        const float4* h4 = (const float4*)hr;
#pragma unroll
        for (int q = 0; q < F / 4; ++q) {
            float4 t = h4[q];
            hv[4 * q + 0] = t.x; hv[4 * q + 1] = t.y;
            hv[4 * q + 2] = t.z; hv[4 * q + 3] = t.w;
        }
    } else if constexpr ((F & 1) == 0) {      // 8B-aligned rows (F=6)
        const float2* h2p = (const float2*)hr;
#pragma unroll
        for (int q = 0; q < F / 2; ++q) {
            float2 t = h2p[q];
            hv[2 * q + 0] = t.x; hv[2 * q + 1] = t.y;
        }
    } else {
#pragma unroll
        for (int f = 0; f < F; ++f) hv[f] = hr[f];
    }

#pragma unroll
    for (int f = 0; f < F; ++f) atomicAdd(oc + f, w * hv[f]);
}

__global__ void bias_relu_kernel(const float* __restrict__ agg,
                                 const float* __restrict__ b,
                                 float* __restrict__ out, long total, int F) {
    long i = (long)blockIdx.x * blockDim.x + threadIdx.x;
    if (i >= total) return;
    float v = agg[i] + b[(int)(i % F)];
    out[i] = v > 0.0f ? v : 0.0f;
}

// Fused layer-3 epilogue: relu(agg3+b3) dot Wl + bl -> sigmoid
__global__ void final_kernel(const float* __restrict__ agg3,
                             const float* __restrict__ b3,
                             const float* __restrict__ Wl,
                             const float* __restrict__ bl,
                             float* __restrict__ out, int n) {
    int i = blockIdx.x * blockDim.x + threadIdx.x;
    if (i >= n) return;
    float v = bl[0];
#pragma unroll
    for (int f = 0; f < 3; ++f) {
        float h = agg3[(long)i * 3 + f] + b3[f];
        h = h > 0.0f ? h : 0.0f;
        v += h * Wl[f];
    }
    out[i] = 1.0f / (1.0f + __expf(-v));
}

// ---------------------------------------------------------------------------
// Launch
// ---------------------------------------------------------------------------
extern "C" void kernel_launch(void* const* d_in, const int* in_sizes, int n_in,
                              void* d_out, int out_size, void* d_ws, size_t ws_size,
                              hipStream_t stream) {
    const float* X  = (const float*)d_in[0];
    const int*   ei = (const int*)d_in[1];   // [2, E], int32 per harness contract
    const float* ew = (const float*)d_in[2];
    const float* W1 = (const float*)d_in[3];
    const float* b1 = (const float*)d_in[4];
    const float* W2 = (const float*)d_in[5];
    const float* b2 = (const float*)d_in[6];
    const float* W3 = (const float*)d_in[7];
    const float* b3 = (const float*)d_in[8];
    const float* Wl = (const float*)d_in[9];
    const float* bl = (const float*)d_in[10];

    const int n = in_sizes[0] / NODE_DIM_IN;
    const int E = in_sizes[2];
    const int* row = ei;
    const int* col = ei + E;

    // Workspace carve-out (256B aligned slices)
    char* p = (char*)d_ws;
    auto carve = [&](size_t bytes) -> float* {
        float* q = (float*)p;
        p += (bytes + 255) & ~(size_t)255;
        return q;
    };
    float* dis   = carve((size_t)n * 4);           // deg -> dis in place
    float* selfn = carve((size_t)n * 4);
    float* nrm   = carve((size_t)E * 4);
    float* h1    = carve((size_t)n * 12 * 4);
    float* agg1  = carve((size_t)n * 12 * 4);
    float* h2    = carve((size_t)n * 6 * 4 + 64);  // +pad: Kp=8 over-read on last row
    float* agg2  = carve((size_t)n * 6 * 4);
    float* h3    = carve((size_t)n * 3 * 4);
    float* agg3  = carve((size_t)n * 3 * 4);

    const int TB = 256;
    const int gn = (n + TB - 1) / TB;
    const int gE = (E + TB - 1) / TB;
    auto gEl = [&](long t) { return (int)((t + TB - 1) / TB); };
    const int ggemm = (n + 63) / 64;               // 4 waves x 16 rows per block

    // --- normalization precompute (shared across layers) ---
    deg_init_kernel<<<gn, TB, 0, stream>>>(dis, n);
    deg_acc_kernel<<<gE, TB, 0, stream>>>(col, ew, dis, E);
    dinv_kernel<<<gn, TB, 0, stream>>>(dis, selfn, n);
    edge_norm_kernel<<<gE, TB, 0, stream>>>(row, col, ew, dis, nrm, E);

    // --- layer 1: K=128, F=12 ---
    wmma_gemm_kernel<128, 12><<<ggemm, 128, 0, stream>>>(X, W1, selfn, h1, agg1, n);
    scatter_kernel<12><<<gE, TB, 0, stream>>>(row, col, nrm, h1, agg1, E);
    bias_relu_kernel<<<gEl((long)n * 12), TB, 0, stream>>>(agg1, b1, h1, (long)n * 12, 12);

    // --- layer 2: K=12, F=6 ---
    wmma_gemm_kernel<12, 6><<<ggemm, 128, 0, stream>>>(h1, W2, selfn, h2, agg2, n);
    scatter_kernel<6><<<gE, TB, 0, stream>>>(row, col, nrm, h2, agg2, E);
    bias_relu_kernel<<<gEl((long)n * 6), TB, 0, stream>>>(agg2, b2, h2, (long)n * 6, 6);

    // --- layer 3: K=6, F=3 ---
    wmma_gemm_kernel<6, 3><<<ggemm, 128, 0, stream>>>(h2, W3, selfn, h3, agg3, n);
    scatter_kernel<3><<<gE, TB, 0, stream>>>(row, col, nrm, h3, agg3, E);

    // --- fused epilogue + final linear + sigmoid ---
    final_kernel<<<gn, TB, 0, stream>>>(agg3, b3, Wl, bl, (float*)d_out, n);
}